// GravityLayer_47674136985717
// MI455X (gfx1250) — compile-verified
//
#include <hip/hip_runtime.h>
#include <hip/hip_bf16.h>
#include <math.h>

// GravityLayer for MI455X (gfx1250): fp32 WMMA (V_WMMA_F32_16X16X4_F32) for the
// pairwise force MLP. Problem is compute-bound (all tensors L2-resident), fp32
// reference => fp32 matrix path. GELU is a branch-free erf approximation so the
// pointwise VALU/TRANS work co-executes with the WMMA pipe (no EXEC churn).

#define NNODES 1024
#define DD 48
#define BB 2
#define SLICES 4
#define WAVES_PER_WG 4
#define H1STRIDE 52                    // padded LDS row stride (floats): kills 2-way bank conflicts, keeps b64 align
#define WAVE_LDS (16 * H1STRIDE + 16)  // h1 staging + 16-float fm slot per wave

typedef float v2f __attribute__((ext_vector_type(2)));
typedef float v8f __attribute__((ext_vector_type(8)));

// Branch-free exact-GELU: erf via Abramowitz-Stegun 7.1.26 (|err| <= 1.5e-7).
// Straight-line VALU + v_rcp_f32 + v_exp_f32; copysignf -> v_bfi_b32.
__device__ __forceinline__ float gelu_exact(float x) {
    float z = fabsf(x) * 0.7071067811865476f;            // |x|/sqrt(2)
    float t = __builtin_amdgcn_rcpf(fmaf(0.3275911f, z, 1.0f));
    float p = fmaf(1.061405429f, t, -1.453152027f);
    p = fmaf(p, t, 1.421413741f);
    p = fmaf(p, t, -0.284496736f);
    p = fmaf(p, t, 0.254829592f);
    p = p * t;
    float e = __expf(-z * z);
    float er = fmaf(-p, e, 1.0f);                        // erf(|x|/sqrt(2))
    er = copysignf(er, x);
    return 0.5f * x * (1.0f + er);
}
__device__ __forceinline__ float softplus_f(float x) {
    return (x > 20.0f) ? x : log1pf(expf(x));
}

// ---------------- kernel 1: per-node mass MLP (48 -> 12 -> 1) ----------------
__global__ void mass_kernel(const float* __restrict__ pos,
                            const float* __restrict__ m_w1, const float* __restrict__ m_b1,
                            const float* __restrict__ m_w2, const float* __restrict__ m_b2,
                            float* __restrict__ mass) {
    int idx = blockIdx.x * blockDim.x + threadIdx.x;
    if (idx >= BB * NNODES) return;
    const float* p = pos + (size_t)idx * DD;
    float pr[DD];
#pragma unroll
    for (int c = 0; c < DD; ++c) pr[c] = p[c];
    float acc = m_b2[0];
#pragma unroll
    for (int k = 0; k < 12; ++k) {
        float s = m_b1[k];
#pragma unroll
        for (int c = 0; c < DD; ++c) s = fmaf(pr[c], m_w1[c * 12 + k], s);
        acc = fmaf(gelu_exact(s), m_w2[k], acc);
    }
    mass[idx] = softplus_f(acc);
}

// ---------------- kernel 2: pairwise force via fp32 WMMA ----------------
// Wave tile = 16 pairs: rows i0..i0+15 against one j. Lane (h=lane>>4, n=lane&15)
// owns pair n, components h*24..h*24+23.
__global__ void __launch_bounds__(128, 1) pair_kernel(
    const float* __restrict__ pos, const float* __restrict__ mass,
    const float* __restrict__ f_w1, const float* __restrict__ f_b1,
    const float* __restrict__ f_w2, const float* __restrict__ f_b2,
    const float* __restrict__ f_w3, const float* __restrict__ f_b3,
    float* __restrict__ force) {
    __shared__ float lds[WAVES_PER_WG * WAVE_LDS];
    const int lane = threadIdx.x & 31;
    const int wave = threadIdx.x >> 5;
    const int h = lane >> 4;
    const int n = lane & 15;
    const int i0 = blockIdx.x * 16;
    const int b = blockIdx.y;
    const int g = blockIdx.z * WAVES_PER_WG + wave;  // global wave slot 0..15
    float* wlds = lds + wave * WAVE_LDS;
    float* wfm = wlds + 16 * H1STRIDE;

    const int i = i0 + n;
    const float* posB = pos + (size_t)b * NNODES * DD;
    const float* massB = mass + (size_t)b * NNODES;

    // pos of my i-row (24 comps), my mass
    float posI[24];
    {
        const float* pi = posB + (size_t)i * DD + h * 24;
#pragma unroll
        for (int c = 0; c < 24; c += 4) {
            float4 q = *(const float4*)(pi + c);
            posI[c] = q.x; posI[c + 1] = q.y; posI[c + 2] = q.z; posI[c + 3] = q.w;
        }
    }
    const float massI = massB[i];

    // layer-1 B fragments (W1 padded to K=4 with a zero row), biases, w3
    float w1b0[3], w1b1[3], b1r[3], b2r[3], w3r[3];
#pragma unroll
    for (int nb = 0; nb < 3; ++nb) {
        int col = nb * 16 + n;
        w1b0[nb] = (2 * h < 3) ? f_w1[(2 * h) * DD + col] : 0.0f;
        w1b1[nb] = (2 * h + 1 < 3) ? f_w1[(2 * h + 1) * DD + col] : 0.0f;
        b1r[nb] = f_b1[col];
        b2r[nb] = f_b2[col];
        w3r[nb] = f_w3[col];
    }
    const float b3 = f_b3[0];

    // layer-2 B fragments held in registers: 12 K-steps x 3 N-blocks x 2
    float w2f[12][3][2];
#pragma unroll
    for (int t = 0; t < 12; ++t)
#pragma unroll
        for (int nb = 0; nb < 3; ++nb) {
            int col = nb * 16 + n;
            w2f[t][nb][0] = f_w2[(4 * t + 2 * h) * DD + col];
            w2f[t][nb][1] = f_w2[(4 * t + 2 * h + 1) * DD + col];
        }

    float F[24];
#pragma unroll
    for (int c = 0; c < 24; ++c) F[c] = 0.0f;

    const int jmax = i0 + 16;  // causal: only j <= i needed; per-row mask below
    for (int j = g; j < jmax; j += 16) {
        int jn = j + 16;
        if (jn < jmax) __builtin_prefetch(posB + (size_t)jn * DD, 0, 0);

        const float mj = massB[j];
        float dispv[24];
        float ss = 0.0f;
        {
            const float* pj = posB + (size_t)j * DD + h * 24;
#pragma unroll
            for (int c = 0; c < 24; c += 4) {
                float4 q = *(const float4*)(pj + c);
                dispv[c] = posI[c] - q.x;
                dispv[c + 1] = posI[c + 1] - q.y;
                dispv[c + 2] = posI[c + 2] - q.z;
                dispv[c + 3] = posI[c + 3] - q.w;
            }
#pragma unroll
            for (int c = 0; c < 24; ++c) ss = fmaf(dispv[c], dispv[c], ss);
        }
        ss += __shfl_xor(ss, 16, 32);  // combine halves -> full ||disp||^2
        const float dist = fmaxf(sqrtf(ss), 0.01f);

        // ---- layer 1: X[16x4] @ W1e[4x48] (X = [dist, m_i, m_j, 0]) ----
        v2f af;                      // A frag: lane holds K=2h, 2h+1 of row n
        af.x = h ? mj : dist;
        af.y = h ? 0.0f : massI;
        v8f acc1[3];
#pragma unroll
        for (int nb = 0; nb < 3; ++nb) {
            v2f bf; bf.x = w1b0[nb]; bf.y = w1b1[nb];
            v8f cz = {};
            acc1[nb] = __builtin_amdgcn_wmma_f32_16x16x4_f32(
                false, af, false, bf, (short)0, cz, false, false);
        }
        // GELU in C/D layout, stage h1[16][48] to LDS for A-layout reload
#pragma unroll
        for (int nb = 0; nb < 3; ++nb)
#pragma unroll
            for (int r = 0; r < 8; ++r) {
                float v = gelu_exact(acc1[nb][r] + b1r[nb]);
                wlds[(r + 8 * h) * H1STRIDE + nb * 16 + n] = v;
            }
        asm volatile("s_wait_dscnt 0" ::: "memory");

        // ---- layer 2: H1[16x48] @ W2[48x48], 12 K-steps x 3 N-blocks ----
        v8f acc2[3];
#pragma unroll
        for (int nb = 0; nb < 3; ++nb) { v8f z = {}; acc2[nb] = z; }
#pragma unroll
        for (int t = 0; t < 12; ++t) {
            v2f av = *(v2f*)&wlds[n * H1STRIDE + 4 * t + 2 * h];  // b64, 8B aligned
#pragma unroll
            for (int nb = 0; nb < 3; ++nb) {
                v2f bf; bf.x = w2f[t][nb][0]; bf.y = w2f[t][nb][1];
                acc2[nb] = __builtin_amdgcn_wmma_f32_16x16x4_f32(
                    false, av, false, bf, (short)0, acc2[nb], false, false);
            }
        }

        // ---- epilogue: fm[row] = b3 + sum_col gelu(h2pre + b2)*w3 ----
#pragma unroll
        for (int r = 0; r < 8; ++r) {
            float p = 0.0f;
#pragma unroll
            for (int nb = 0; nb < 3; ++nb)
                p = fmaf(gelu_exact(acc2[nb][r] + b2r[nb]), w3r[nb], p);
            p += __shfl_xor(p, 1, 32);
            p += __shfl_xor(p, 2, 32);
            p += __shfl_xor(p, 4, 32);
            p += __shfl_xor(p, 8, 32);     // reduce across N-lanes within half
            if (n == r) wfm[r + 8 * h] = p + b3;  // row = r + 8h
        }
        asm volatile("s_wait_dscnt 0" ::: "memory");
        float fmv = wfm[n];
        if (j > i) fmv = 0.0f;  // causal mask (j == i gives direction == 0 anyway)
        const float sc = fmv / dist;
#pragma unroll
        for (int c = 0; c < 24; ++c) F[c] = fmaf(sc, dispv[c], F[c]);
    }

    float* fout = force + ((size_t)b * NNODES + i) * DD + h * 24;
#pragma unroll
    for (int c = 0; c < 24; ++c) unsafeAtomicAdd(&fout[c], F[c]);
}

// ---------------- kernel 3: integrator ----------------
__global__ void finalize_kernel(const float* __restrict__ pos, const float* __restrict__ vel,
                                const float* __restrict__ force, const float* __restrict__ mass,
                                const float* __restrict__ dampp, const float* __restrict__ dtp,
                                float* __restrict__ out) {
    int idx = blockIdx.x * blockDim.x + threadIdx.x;
    const int TOT = BB * NNODES * DD;
    if (idx >= TOT) return;
    const float damping = dampp[0], dt = dtp[0];
    int node = idx / DD;
    float acc = force[idx] / (mass[node] + 0.1f);
    float nv = fmaf(dt, acc, damping * vel[idx]);
    float np = fmaf(dt, nv, pos[idx]);
    out[idx] = np;        // new_pos
    out[TOT + idx] = nv;  // new_vel
}

extern "C" void kernel_launch(void* const* d_in, const int* in_sizes, int n_in,
                              void* d_out, int out_size, void* d_ws, size_t ws_size,
                              hipStream_t stream) {
    const float* pos = (const float*)d_in[0];
    const float* vel = (const float*)d_in[1];
    const float* m_w1 = (const float*)d_in[2];
    const float* m_b1 = (const float*)d_in[3];
    const float* m_w2 = (const float*)d_in[4];
    const float* m_b2 = (const float*)d_in[5];
    const float* f_w1 = (const float*)d_in[6];
    const float* f_b1 = (const float*)d_in[7];
    const float* f_w2 = (const float*)d_in[8];
    const float* f_b2 = (const float*)d_in[9];
    const float* f_w3 = (const float*)d_in[10];
    const float* f_b3 = (const float*)d_in[11];
    const float* damp = (const float*)d_in[12];
    const float* dt = (const float*)d_in[13];
    float* outf = (float*)d_out;

    float* mass = (float*)d_ws;                 // B*N floats
    float* force = mass + BB * NNODES;          // B*N*D floats

    hipMemsetAsync(force, 0, (size_t)BB * NNODES * DD * sizeof(float), stream);
    mass_kernel<<<(BB * NNODES + 255) / 256, 256, 0, stream>>>(pos, m_w1, m_b1, m_w2, m_b2, mass);
    dim3 grid(NNODES / 16, BB, SLICES);
    pair_kernel<<<grid, WAVES_PER_WG * 32, 0, stream>>>(pos, mass, f_w1, f_b1, f_w2, f_b2,
                                                        f_w3, f_b3, force);
    finalize_kernel<<<(BB * NNODES * DD + 255) / 256, 256, 0, stream>>>(pos, vel, force, mass,
                                                                        damp, dt, outf);
}